// OnlineLRUCell_85804856639686
// MI455X (gfx1250) — compile-verified
//
#include <hip/hip_runtime.h>

#define H_DIM 1024
#define D_DIM 1024
#define BATCH 16

typedef float v2f __attribute__((ext_vector_type(2)));
typedef float v8f __attribute__((ext_vector_type(8)));

// ---------------------------------------------------------------------------
// Kernel 1: Cr = x @ B_real^T, Ci = x @ B_img^T  via V_WMMA_F32_16X16X4_F32.
// One wave (32 lanes) per 16-wide H tile; M=16 batch fits one tile exactly.
// f32 A (16x4) fragment: lanes 0-15 hold K=0,1; lanes 16-31 hold K=2,3 (float2).
// f32 B (4x16) fragment: lane = N (+16 for K-half), float2 of consecutive K.
// Both fragments are contiguous float2 loads from row-major x / B matrices.
// ---------------------------------------------------------------------------
__global__ __launch_bounds__(32) void lru_gemm_f32_wmma(
    const float* __restrict__ x,   // (16, D)
    const float* __restrict__ Br,  // (H, D)
    const float* __restrict__ Bi,  // (H, D)
    float* __restrict__ Cr,        // (16, H) ws
    float* __restrict__ Ci)        // (16, H) ws
{
    const int lane  = threadIdx.x;          // 0..31, EXEC all ones
    const int n0    = blockIdx.x << 4;      // H tile base
    const int mn    = lane & 15;            // A row / B column within tile
    const int khalf = (lane >> 4) << 1;     // 0 or 2

    const float* xa = x  + mn * D_DIM + khalf;
    const float* br = Br + (size_t)(n0 + mn) * D_DIM + khalf;
    const float* bi = Bi + (size_t)(n0 + mn) * D_DIM + khalf;

    v8f accr = {};
    v8f acci = {};
    for (int k = 0; k < D_DIM; k += 4) {
        v2f a  = *(const v2f*)(xa + k);
        v2f b0 = *(const v2f*)(br + k);
        v2f b1 = *(const v2f*)(bi + k);
        accr = __builtin_amdgcn_wmma_f32_16x16x4_f32(
            false, a, false, b0, (short)0, accr, false, false);
        acci = __builtin_amdgcn_wmma_f32_16x16x4_f32(
            false, a, false, b1, (short)0, acci, false, false);
    }

    // C/D layout: VGPR r, lanes 0-15 -> M=r, N=lane; lanes 16-31 -> M=r+8.
    const int mbase = (lane >> 4) << 3;
#pragma unroll
    for (int r = 0; r < 8; ++r) {
        const int m = mbase + r;
        Cr[m * H_DIM + n0 + mn] = accr[r];
        Ci[m * H_DIM + n0 + mn] = acci[r];
    }
}

// ---------------------------------------------------------------------------
// Kernel 2: pointwise h_t / gl / gg updates (B*H = 16K threads). Lambda
// recomputed inline (5 transcendentals per thread; negligible).
// Outputs are interleaved complex float pairs.
// ---------------------------------------------------------------------------
__global__ __launch_bounds__(256) void lru_pointwise(
    const float* __restrict__ h_re,  const float* __restrict__ h_im,
    const float* __restrict__ gl_re, const float* __restrict__ gl_im,
    const float* __restrict__ gg_re, const float* __restrict__ gg_im,
    const float* __restrict__ nu_log, const float* __restrict__ theta_log,
    const float* __restrict__ gamma_log,
    const float* __restrict__ Cr, const float* __restrict__ Ci,
    float* __restrict__ outHt, float* __restrict__ outGl,
    float* __restrict__ outGg)
{
    const int idx = blockIdx.x * blockDim.x + threadIdx.x;  // 0..16383
    const int h   = idx & (H_DIM - 1);

    const float el = __expf(-__expf(nu_log[h]));
    const float th = __expf(theta_log[h]);
    const float lr = el * __cosf(th);
    const float li = el * __sinf(th);
    const float eg = __expf(gamma_log[h]);

    const float cr = Cr[idx];
    const float ci = Ci[idx];
    const float hr = h_re[idx], hi = h_im[idx];

    // h_t = Lam*h + e^gamma * (Cr + i Ci)
    outHt[2 * idx]     = lr * hr - li * hi + eg * cr;
    outHt[2 * idx + 1] = lr * hi + li * hr + eg * ci;

    // gl = Lam*gl_prev + h
    const float glr = gl_re[idx], gli = gl_im[idx];
    outGl[2 * idx]     = lr * glr - li * gli + hr;
    outGl[2 * idx + 1] = lr * gli + li * glr + hi;

    // gg = Lam*gg_prev + (Cr + i Ci)
    const float ggr = gg_re[idx], ggi = gg_im[idx];
    outGg[2 * idx]     = lr * ggr - li * ggi + cr;
    outGg[2 * idx + 1] = lr * ggi + li * ggr + ci;
}

// ---------------------------------------------------------------------------
// Kernel 3: gB streamer — the bandwidth roofline (~256 MB HBM traffic).
// Each 256-thread block covers 4 full D-rows (4096 elements); Lambda/gamma for
// those 4 h-values computed once into LDS. 16 elements per thread, float4
// global loads and interleaved-complex float4 stores.
//   gB_out = Lam * gB_prev + gamma_log[h] * x[b,d]   (added term is real)
// ---------------------------------------------------------------------------
__global__ __launch_bounds__(256) void lru_gB_stream(
    const float* __restrict__ x,
    const float* __restrict__ gB_re, const float* __restrict__ gB_im,
    const float* __restrict__ nu_log, const float* __restrict__ theta_log,
    const float* __restrict__ gamma_log,
    float* __restrict__ outGB)
{
    __shared__ float s_lr[4], s_li[4], s_gl[4];

    const int t       = threadIdx.x;
    const int bh_base = blockIdx.x << 2;   // 4 (b,h) rows per block

    if (t < 4) {
        const int h  = (bh_base + t) & (H_DIM - 1);
        const float el = __expf(-__expf(nu_log[h]));
        const float th = __expf(theta_log[h]);
        s_lr[t] = el * __cosf(th);
        s_li[t] = el * __sinf(th);
        s_gl[t] = gamma_log[h];
    }
    __syncthreads();

    const int r  = t >> 6;            // row within block: 0..3
    const int d0 = (t & 63) << 4;     // 16 consecutive d per thread
    const int bh = bh_base + r;
    const int b  = bh >> 10;          // bh / H

    const float lr = s_lr[r];
    const float li = s_li[r];
    const float gl = s_gl[r];

    const size_t base = (size_t)bh * D_DIM + d0;
    const float4* gr4 = (const float4*)(gB_re + base);
    const float4* gi4 = (const float4*)(gB_im + base);
    const float4* x4  = (const float4*)(x + (size_t)b * D_DIM + d0);
    float4* o4        = (float4*)(outGB + 2 * base);

#pragma unroll
    for (int i = 0; i < 4; ++i) {
        const float4 gr = gr4[i];
        const float4 gi = gi4[i];
        const float4 xv = x4[i];
        float4 o0, o1;
        o0.x = lr * gr.x - li * gi.x + gl * xv.x;  o0.y = lr * gi.x + li * gr.x;
        o0.z = lr * gr.y - li * gi.y + gl * xv.y;  o0.w = lr * gi.y + li * gr.y;
        o1.x = lr * gr.z - li * gi.z + gl * xv.z;  o1.y = lr * gi.z + li * gr.z;
        o1.z = lr * gr.w - li * gi.w + gl * xv.w;  o1.w = lr * gi.w + li * gr.w;
        o4[2 * i]     = o0;
        o4[2 * i + 1] = o1;
    }
}

// ---------------------------------------------------------------------------
// Host launcher. Input order (setup_inputs): x, h_re, h_im, gl_re, gl_im,
// gg_re, gg_im, gB_re, gB_im, nu_log, theta_log, gamma_log, B_real, B_img.
// Output: h_t | gl | gg | gB, each complex64 interleaved (re, im) floats.
// ---------------------------------------------------------------------------
extern "C" void kernel_launch(void* const* d_in, const int* in_sizes, int n_in,
                              void* d_out, int out_size, void* d_ws, size_t ws_size,
                              hipStream_t stream)
{
    const float* x         = (const float*)d_in[0];
    const float* h_re      = (const float*)d_in[1];
    const float* h_im      = (const float*)d_in[2];
    const float* gl_re     = (const float*)d_in[3];
    const float* gl_im     = (const float*)d_in[4];
    const float* gg_re     = (const float*)d_in[5];
    const float* gg_im     = (const float*)d_in[6];
    const float* gB_re     = (const float*)d_in[7];
    const float* gB_im     = (const float*)d_in[8];
    const float* nu_log    = (const float*)d_in[9];
    const float* theta_log = (const float*)d_in[10];
    const float* gamma_log = (const float*)d_in[11];
    const float* B_real    = (const float*)d_in[12];
    const float* B_img     = (const float*)d_in[13];

    float* out   = (float*)d_out;
    float* outHt = out;                                   // 2*B*H floats
    float* outGl = out + 2 * BATCH * H_DIM;
    float* outGg = out + 4 * BATCH * H_DIM;
    float* outGB = out + 6 * BATCH * H_DIM;               // 2*B*H*D floats

    float* Cr = (float*)d_ws;                             // (B,H)
    float* Ci = Cr + BATCH * H_DIM;                       // (B,H)

    // Big bandwidth-bound streamer first (independent of the GEMM).
    {
        const int nblocks = (BATCH * H_DIM) / 4;          // 4096
        lru_gB_stream<<<nblocks, 256, 0, stream>>>(
            x, gB_re, gB_im, nu_log, theta_log, gamma_log, outGB);
    }

    // WMMA GEMM: Cr/Ci into workspace.
    lru_gemm_f32_wmma<<<H_DIM / 16, 32, 0, stream>>>(x, B_real, B_img, Cr, Ci);

    // Pointwise h_t / gl / gg consuming the GEMM results.
    lru_pointwise<<<(BATCH * H_DIM) / 256, 256, 0, stream>>>(
        h_re, h_im, gl_re, gl_im, gg_re, gg_im,
        nu_log, theta_log, gamma_log, Cr, Ci,
        outHt, outGl, outGg);
}